// MoM_66391604462090
// MI455X (gfx1250) — compile-verified
//
#include <hip/hip_runtime.h>
#include <hip/hip_bf16.h>

// ---------------------------------------------------------------------------
// Problem constants (match reference)
// ---------------------------------------------------------------------------
#define Bsz   64
#define Ntok  256
#define Dm    1024
#define E_T   4
#define E_C   4
#define NE    8          // E = E_T + E_C
#define HT_   1024       // 4*N
#define HC_   4096       // 4*D
#define HM_   4096       // 4*D
#define SIXD  (6*Dm)
#define BND   (Bsz*Ntok*Dm)

typedef __attribute__((ext_vector_type(2))) float v2f;
typedef __attribute__((ext_vector_type(8))) float v8f;

// ---------------------------------------------------------------------------
// WMMA helper: D(16x16,f32) += A(16x4,f32) x B(4x16,f32)
// 32-bit A layout (ISA 7.12.2): lane L holds row M=L&15; VGPR v holds
// K = v + 2*(L>>4).  B mirrors with N=L&15.  C/D: lane L col N=L&15,
// VGPR v row M = v + 8*(L>>4).
// ---------------------------------------------------------------------------
__device__ __forceinline__ v8f wmma4(float ax, float ay, float bx, float by, v8f c) {
  v2f a = {ax, ay};
  v2f b = {bx, by};
  return __builtin_amdgcn_wmma_f32_16x16x4_f32(false, a, false, b, (short)0, c, false, false);
}

__device__ __forceinline__ float gelu_tanh(float x) {
  const float k0 = 0.7978845608028654f;   // sqrt(2/pi)
  const float k1 = 0.044715f;
  float u = k0 * (x + k1 * x * x * x);
  return 0.5f * x * (1.0f + tanhf(u));
}

__device__ __forceinline__ float silu_f(float x) {
  return x / (1.0f + __expf(-x));
}

// CDNA5 async copy: 16 bytes global -> LDS per lane, tracked by ASYNCcnt.
__device__ __forceinline__ void async_g2l_b128(unsigned lds_byte_addr,
                                               const float* gaddr) {
  asm volatile("global_load_async_to_lds_b128 %0, %1, off"
               :: "v"(lds_byte_addr), "v"(gaddr) : "memory");
}
__device__ __forceinline__ void async_wait0() {
  asm volatile("s_wait_asynccnt 0x0" ::: "memory");
}

// ---------------------------------------------------------------------------
// 1) silu(c) precompute
// ---------------------------------------------------------------------------
__global__ __launch_bounds__(256) void k_silu(const float* __restrict__ c,
                                              float* __restrict__ sc) {
  int i = blockIdx.x * 256 + threadIdx.x;
  if (i < Bsz * Dm) sc[i] = silu_f(c[i]);
}

// ---------------------------------------------------------------------------
// 2) mod = silu(c) @ w_ada + b_ada   (64 x 6144, K=1024) — WMMA f32
// ---------------------------------------------------------------------------
__global__ __launch_bounds__(128) void k_ada(const float* __restrict__ sc,
                                             const float* __restrict__ w_ada,
                                             const float* __restrict__ b_ada,
                                             float* __restrict__ mod) {
  const int ntn  = SIXD / 16;                       // 384 col tiles
  int wave = blockIdx.x * (blockDim.x >> 5) + (threadIdx.x >> 5);
  int lane = threadIdx.x & 31;
  int r  = lane & 15;
  int kh = lane >> 4;
  int m0 = (wave / ntn) * 16;
  int n0 = (wave % ntn) * 16;
  v8f acc = {};
  const int rowA = m0 + r;
  for (int k0 = 0; k0 < Dm; k0 += 4) {
    int ka = k0 + 2 * kh;
    float ax = sc[rowA * Dm + ka];
    float ay = sc[rowA * Dm + ka + 1];
    float bx = w_ada[ka * SIXD + n0 + r];
    float by = w_ada[(ka + 1) * SIXD + n0 + r];
    if (k0 + 4 < Dm)
      __builtin_prefetch(&w_ada[(ka + 4) * SIXD + n0 + r], 0, 1);
    acc = wmma4(ax, ay, bx, by, acc);
  }
#pragma unroll
  for (int v = 0; v < 8; ++v) {
    int row = m0 + v + 8 * kh;
    int col = n0 + r;
    mod[row * SIXD + col] = acc[v] + b_ada[col];
  }
}

// ---------------------------------------------------------------------------
// 3) h = modulate(LN(x), s_msa, sc_msa)
// ---------------------------------------------------------------------------
__global__ __launch_bounds__(256) void k_lnmod(const float* __restrict__ x,
                                               const float* __restrict__ mod,
                                               float* __restrict__ h) {
  __shared__ float red[256];
  int row = blockIdx.x;                 // b*N + n
  int b   = row / Ntok;
  int tid = threadIdx.x;
  const float* xr = x + (size_t)row * Dm;
  float v[4]; float s = 0.f;
#pragma unroll
  for (int i = 0; i < 4; ++i) { v[i] = xr[tid + i * 256]; s += v[i]; }
  red[tid] = s; __syncthreads();
  for (int off = 128; off > 0; off >>= 1) {
    if (tid < off) red[tid] += red[tid + off];
    __syncthreads();
  }
  float mu = red[0] * (1.0f / Dm);
  __syncthreads();
  float vs = 0.f;
#pragma unroll
  for (int i = 0; i < 4; ++i) { float d = v[i] - mu; vs += d * d; }
  red[tid] = vs; __syncthreads();
  for (int off = 128; off > 0; off >>= 1) {
    if (tid < off) red[tid] += red[tid + off];
    __syncthreads();
  }
  float rs = rsqrtf(red[0] * (1.0f / Dm) + 1e-6f);
  const float* mb = mod + (size_t)b * SIXD;       // [0]=s_msa  [D]=sc_msa
#pragma unroll
  for (int i = 0; i < 4; ++i) {
    int d = tid + i * 256;
    float nrm = (v[i] - mu) * rs;
    h[(size_t)row * Dm + d] = nrm * (1.0f + mb[Dm + d]) + mb[d];
  }
}

// ---------------------------------------------------------------------------
// 4) r_in = mean over tokens of h
// ---------------------------------------------------------------------------
__global__ __launch_bounds__(256) void k_rin(const float* __restrict__ h,
                                             float* __restrict__ rin) {
  int b = blockIdx.x;
  int d = blockIdx.y * 256 + threadIdx.x;
  const float* hp = h + (size_t)b * Ntok * Dm + d;
  float s = 0.f;
  for (int n = 0; n < Ntok; ++n) s += hp[(size_t)n * Dm];
  rin[b * Dm + d] = s * (1.0f / Ntok);
}

// ---------------------------------------------------------------------------
// 5) router: logits -> softmax -> top-2 -> combine weights (+ aux inputs)
// ---------------------------------------------------------------------------
__global__ __launch_bounds__(256) void k_router(const float* __restrict__ rin,
                                                const float* __restrict__ w_router,
                                                float* __restrict__ combine,
                                                float* __restrict__ probs_out,
                                                int* __restrict__ top1) {
  __shared__ float red[256];
  __shared__ float logits[NE];
  int b = blockIdx.x, tid = threadIdx.x;
  float acc[NE];
#pragma unroll
  for (int e = 0; e < NE; ++e) acc[e] = 0.f;
  for (int d = tid; d < Dm; d += 256) {
    float v = rin[b * Dm + d];
#pragma unroll
    for (int e = 0; e < NE; ++e) acc[e] += v * w_router[d * NE + e];
  }
  for (int e = 0; e < NE; ++e) {
    red[tid] = acc[e]; __syncthreads();
    for (int off = 128; off > 0; off >>= 1) {
      if (tid < off) red[tid] += red[tid + off];
      __syncthreads();
    }
    if (tid == 0) logits[e] = red[0];
    __syncthreads();
  }
  if (tid == 0) {
    float mx = logits[0];
    for (int e = 1; e < NE; ++e) mx = fmaxf(mx, logits[e]);
    float p[NE]; float sum = 0.f;
    for (int e = 0; e < NE; ++e) { p[e] = __expf(logits[e] - mx); sum += p[e]; }
    float inv = 1.0f / sum;
    for (int e = 0; e < NE; ++e) p[e] *= inv;
    int i0 = 0;
    for (int e = 1; e < NE; ++e) if (p[e] > p[i0]) i0 = e;
    int i1 = (i0 == 0) ? 1 : 0;
    for (int e = 0; e < NE; ++e) if (e != i0 && p[e] > p[i1]) i1 = e;
    float t0 = p[i0], t1 = p[i1], s = t0 + t1;
    for (int e = 0; e < NE; ++e) combine[b * NE + e] = 0.f;
    combine[b * NE + i0] = t0 / s;
    combine[b * NE + i1] = t1 / s;
    for (int e = 0; e < NE; ++e) probs_out[b * NE + e] = p[e];
    top1[b] = i0;
  }
}

// ---------------------------------------------------------------------------
// 6/7/9) fused expert/MLP FFN:  O = W2^T gelu(W1^T X + b1) + b2
//   MODE 0: token-mixer   rows=(b, 16 d's),    K1=256,  H=1024, K2=256
//   MODE 1: channel-mixer rows=(b, 16 tokens), K1=1024, H=4096, K2=1024
//   MODE 2: dense MLP     rows=(b, 16 tokens), K1=1024, H=4096, K2=1024
// X row-block staged into LDS once via global_load_async_to_lds_b128
// (ASYNCcnt), hidden processed in 1024-wide LDS passes (XOR bank swizzle),
// stage-2 f32 accumulators persist in VGPRs across passes.  One A-fragment
// (ds_load_b64) is shared by 4 WMMA column tiles per k-step.
// ---------------------------------------------------------------------------
template <int MODE>
__global__ __launch_bounds__(256) void k_ffn(const float* __restrict__ Xin,
                                             const float* __restrict__ W1,
                                             const float* __restrict__ B1,
                                             const float* __restrict__ W2,
                                             const float* __restrict__ B2,
                                             const float* __restrict__ combine,
                                             const float* __restrict__ mod,
                                             const float* __restrict__ xmid,
                                             float* __restrict__ Out) {
  constexpr int K1 = (MODE == 0) ? Ntok : Dm;
  constexpr int H  = (MODE == 0) ? HT_  : HC_;
  constexpr int K2 = (MODE == 0) ? Ntok : Dm;
  constexpr int HCH = 1024;
  constexpr int NPASS = H / HCH;
  constexpr int NW = 8;                      // waves per block
  constexpr int T1 = HCH / 16 / NW;          // stage-1 tiles per wave per pass (8)
  constexpr int T2 = K2 / 16 / NW;           // stage-2 tiles per wave (2 or 8)
  constexpr int G1 = 4;                      // stage-1 tiles sharing one A frag
  constexpr int G2 = (T2 < 4) ? T2 : 4;      // stage-2 tiles sharing one A frag

  __shared__ float hid[16 * HCH];            // 64 KB, swizzled
  __shared__ float xs[16 * K1];              // X row-block (16/64 KB)

  int blk = blockIdx.x;
  int b, rb, e;
  if (MODE == 0) {
    b = blk / ((Dm / 16) * E_T);
    int rem = blk % ((Dm / 16) * E_T);
    rb = rem / E_T; e = rem % E_T;
  } else if (MODE == 1) {
    b = blk / ((Ntok / 16) * E_C);
    int rem = blk % ((Ntok / 16) * E_C);
    rb = rem / E_C; e = rem % E_C;
  } else {
    b = blk / (Ntok / 16); rb = blk % (Ntok / 16); e = 0;
  }

  float w = 1.0f;
  if (MODE == 0) w = combine[b * NE + e];
  if (MODE == 1) w = combine[b * NE + E_T + e];
  if (w == 0.0f) return;                     // expert not selected: exact skip

  const float* W1e = W1 + (size_t)e * K1 * H;
  const float* B1e = B1 + (size_t)e * H;
  const float* W2e = W2 + (size_t)e * H * K2;
  const float* B2e = B2 + (size_t)e * K2;

  int wid  = threadIdx.x >> 5;
  int lane = threadIdx.x & 31;
  int r  = lane & 15;
  int kh = lane >> 4;

  const size_t xbase = (MODE == 0)
      ? (size_t)b * Ntok * Dm + (size_t)rb * 16         // X[r][k] = [k*D + r]
      : ((size_t)b * Ntok + rb * 16) * Dm;              // X[r][k] = [r*D + k]

  // ---- async-stage X row-block into LDS ----------------------------------
  // MODE 0: k-major xs[k*16 + r]   (strided global gather -> contiguous b128)
  // MODE 1/2: row-major + XOR swizzle xs[row*K1 + (col ^ 4*row)]
  {
    const int chunks = 16 * K1 / 4;          // 16B chunks
    for (int i = threadIdx.x; i < chunks; i += 256) {
      unsigned loff; const float* g;
      if (MODE == 0) {
        int k = i >> 2, rg = (i & 3) << 2;
        loff = (unsigned)(size_t)&xs[k * 16 + rg];
        g = Xin + xbase + (size_t)k * Dm + rg;
      } else {
        int row = i >> 8, col = (i & 255) << 2;
        loff = (unsigned)(size_t)&xs[row * K1 + (col ^ ((row & 15) << 2))];
        g = Xin + xbase + (size_t)row * Dm + col;
      }
      async_g2l_b128(loff, g);
    }
    async_wait0();
    __syncthreads();
  }

  const int sw = r << 2;                      // XOR swizzle key for this lane

  v8f acc2[T2];
#pragma unroll
  for (int t = 0; t < T2; ++t) acc2[t] = (v8f){};

  for (int pass = 0; pass < NPASS; ++pass) {
    const int hg0 = pass * HCH;
    // ---- stage 1: hidden chunk -> LDS (A shared across G1 tiles) ----------
#pragma unroll 1
    for (int g = 0; g < T1; g += G1) {
      v8f a1[G1];
#pragma unroll
      for (int j = 0; j < G1; ++j) a1[j] = (v8f){};
      for (int k0 = 0; k0 < K1; k0 += 4) {
        int ka = k0 + 2 * kh;
        v2f av;
        if (MODE == 0) {
          av.x = xs[ka * 16 + r];
          av.y = xs[(ka + 1) * 16 + r];
        } else {
          av = *reinterpret_cast<const v2f*>(&xs[r * K1 + (ka ^ sw)]);
        }
        if (k0 + 4 < K1)
          __builtin_prefetch(&W1e[(size_t)(ka + 4) * H + hg0 + (wid * T1 + g) * 16 + r], 0, 1);
#pragma unroll
        for (int j = 0; j < G1; ++j) {
          int ng = hg0 + (wid * T1 + g + j) * 16 + r;
          float bx = W1e[(size_t)ka * H + ng];
          float by = W1e[(size_t)(ka + 1) * H + ng];
          a1[j] = wmma4(av.x, av.y, bx, by, a1[j]);
        }
      }
#pragma unroll
      for (int j = 0; j < G1; ++j) {
        int n0l  = (wid * T1 + g + j) * 16;
        float bias = B1e[hg0 + n0l + r];
#pragma unroll
        for (int v = 0; v < 8; ++v) {
          int rr = v + 8 * kh;
          hid[rr * HCH + ((n0l + r) ^ ((rr & 15) << 2))] = gelu_tanh(a1[j][v] + bias);
        }
      }
    }
    __syncthreads();
    // ---- stage 2: consume chunk (A shared across G2 tiles) ----------------
#pragma unroll 1
    for (int g = 0; g < T2; g += G2) {
      for (int kl = 0; kl < HCH; kl += 4) {
        int ka  = kl + 2 * kh;
        int kga = hg0 + ka;
        v2f av = *reinterpret_cast<const v2f*>(&hid[r * HCH + (ka ^ sw)]);
        if (kl + 4 < HCH)
          __builtin_prefetch(&W2e[(size_t)(kga + 4) * K2 + (wid * T2 + g) * 16 + r], 0, 1);
#pragma unroll
        for (int j = 0; j < G2; ++j) {
          int nc = (wid * T2 + g + j) * 16 + r;
          float bx = W2e[(size_t)kga * K2 + nc];
          float by = W2e[(size_t)(kga + 1) * K2 + nc];
          acc2[g + j] = wmma4(av.x, av.y, bx, by, acc2[g + j]);
        }
      }
    }
    __syncthreads();
  }

  // ---- epilogue -----------------------------------------------------------
#pragma unroll 1
  for (int t = 0; t < T2; ++t) {
    int nc = (wid * T2 + t) * 16 + r;
#pragma unroll
    for (int v = 0; v < 8; ++v) {
      int rr  = v + 8 * kh;
      float val = acc2[t][v] + B2e[nc];
      if (MODE == 0) {
        int d = rb * 16 + rr;                 // oe is (d, n); out[b,n,d] += w*oe
        atomicAdd(&Out[((size_t)b * Ntok + nc) * Dm + d], w * val);
      } else if (MODE == 1) {
        int tok = rb * 16 + rr;
        atomicAdd(&Out[((size_t)b * Ntok + tok) * Dm + nc], w * val);
      } else {
        size_t row = (size_t)b * Ntok + rb * 16 + rr;
        float g = mod[(size_t)b * SIXD + 5 * Dm + nc];  // g_mlp
        Out[row * Dm + nc] = xmid[row * Dm + nc] + g * val;
      }
    }
  }
}

// ---------------------------------------------------------------------------
// 8) x_mid = x + g_msa*mix ;  h2 = modulate(LN(x_mid), s_mlp, sc_mlp)
// ---------------------------------------------------------------------------
__global__ __launch_bounds__(256) void k_resid_ln(const float* __restrict__ x,
                                                  const float* __restrict__ mix,
                                                  const float* __restrict__ mod,
                                                  float* __restrict__ xmid,
                                                  float* __restrict__ h2) {
  __shared__ float red[256];
  int row = blockIdx.x;
  int b   = row / Ntok;
  int tid = threadIdx.x;
  const float* mb = mod + (size_t)b * SIXD;
  float v[4]; float s = 0.f;
#pragma unroll
  for (int i = 0; i < 4; ++i) {
    int d = tid + i * 256;
    float xv = x[(size_t)row * Dm + d] + mb[2 * Dm + d] * mix[(size_t)row * Dm + d];
    v[i] = xv; s += xv;
    xmid[(size_t)row * Dm + d] = xv;
  }
  red[tid] = s; __syncthreads();
  for (int off = 128; off > 0; off >>= 1) {
    if (tid < off) red[tid] += red[tid + off];
    __syncthreads();
  }
  float mu = red[0] * (1.0f / Dm);
  __syncthreads();
  float vs = 0.f;
#pragma unroll
  for (int i = 0; i < 4; ++i) { float d = v[i] - mu; vs += d * d; }
  red[tid] = vs; __syncthreads();
  for (int off = 128; off > 0; off >>= 1) {
    if (tid < off) red[tid] += red[tid + off];
    __syncthreads();
  }
  float rs = rsqrtf(red[0] * (1.0f / Dm) + 1e-6f);
#pragma unroll
  for (int i = 0; i < 4; ++i) {
    int d = tid + i * 256;
    float nrm = (v[i] - mu) * rs;
    h2[(size_t)row * Dm + d] = nrm * (1.0f + mb[4 * Dm + d]) + mb[3 * Dm + d];
  }
}

// ---------------------------------------------------------------------------
// 10) aux = E * sum_e mean_b(probs)[e] * mean_b(onehot(top1))[e]
// ---------------------------------------------------------------------------
__global__ __launch_bounds__(64) void k_aux(const float* __restrict__ probs,
                                            const int* __restrict__ top1,
                                            float* __restrict__ out_aux) {
  __shared__ float sp[NE], sf[NE];
  if (threadIdx.x < NE) { sp[threadIdx.x] = 0.f; sf[threadIdx.x] = 0.f; }
  __syncthreads();
  if (threadIdx.x < Bsz) {
    int b = threadIdx.x;
    for (int e = 0; e < NE; ++e) atomicAdd(&sp[e], probs[b * NE + e]);
    atomicAdd(&sf[top1[b]], 1.0f);
  }
  __syncthreads();
  if (threadIdx.x == 0) {
    float a = 0.f;
    for (int e = 0; e < NE; ++e)
      a += (sp[e] * (1.0f / Bsz)) * (sf[e] * (1.0f / Bsz));
    out_aux[0] = (float)NE * a;
  }
}

// ---------------------------------------------------------------------------
// launch
// ---------------------------------------------------------------------------
extern "C" void kernel_launch(void* const* d_in, const int* in_sizes, int n_in,
                              void* d_out, int out_size, void* d_ws, size_t ws_size,
                              hipStream_t stream) {
  const float* x        = (const float*)d_in[0];
  const float* c        = (const float*)d_in[1];
  const float* w_ada    = (const float*)d_in[2];
  const float* b_ada    = (const float*)d_in[3];
  const float* w_router = (const float*)d_in[4];
  const float* tm_w1    = (const float*)d_in[5];
  const float* tm_b1    = (const float*)d_in[6];
  const float* tm_w2    = (const float*)d_in[7];
  const float* tm_b2    = (const float*)d_in[8];
  const float* cm_w1    = (const float*)d_in[9];
  const float* cm_b1    = (const float*)d_in[10];
  const float* cm_w2    = (const float*)d_in[11];
  const float* cm_b2    = (const float*)d_in[12];
  const float* mlp_w1   = (const float*)d_in[13];
  const float* mlp_b1   = (const float*)d_in[14];
  const float* mlp_w2   = (const float*)d_in[15];
  const float* mlp_b2   = (const float*)d_in[16];

  float* out = (float*)d_out;                       // [B*N*D] x  +  [1] aux

  // workspace carve-up (floats); requires ~271 MB
  float* ws   = (float*)d_ws;
  float* sc   = ws;                 ws += Bsz * Dm;          // silu(c)
  float* mod  = ws;                 ws += Bsz * SIXD;
  float* h    = ws;                 ws += BND;
  float* rin  = ws;                 ws += Bsz * Dm;
  float* comb = ws;                 ws += Bsz * NE;
  float* prob = ws;                 ws += Bsz * NE;
  int*   top1 = (int*)ws;           ws += Bsz;
  float* mix  = ws;                 ws += BND;
  float* xmid = ws;                 ws += BND;
  float* h2   = ws;                 /* ws += BND; */

  k_silu<<<(Bsz * Dm) / 256, 256, 0, stream>>>(c, sc);
  k_ada<<<(4 * (SIXD / 16)) / 4, 128, 0, stream>>>(sc, w_ada, b_ada, mod);
  k_lnmod<<<Bsz * Ntok, 256, 0, stream>>>(x, mod, h);
  k_rin<<<dim3(Bsz, Dm / 256), 256, 0, stream>>>(h, rin);
  k_router<<<Bsz, 256, 0, stream>>>(rin, w_router, comb, prob, top1);

  hipMemsetAsync(mix, 0, (size_t)BND * sizeof(float), stream);

  // token-mixer experts (exact skip on combine==0 inside)
  k_ffn<0><<<Bsz * (Dm / 16) * E_T, 256, 0, stream>>>(
      h, tm_w1, tm_b1, tm_w2, tm_b2, comb, mod, nullptr, mix);
  // channel-mixer experts
  k_ffn<1><<<Bsz * (Ntok / 16) * E_C, 256, 0, stream>>>(
      h, cm_w1, cm_b1, cm_w2, cm_b2, comb, mod, nullptr, mix);

  k_resid_ln<<<Bsz * Ntok, 256, 0, stream>>>(x, mix, mod, xmid, h2);

  // dense MLP writes final x (with residual + g_mlp gate)
  k_ffn<2><<<Bsz * (Ntok / 16), 256, 0, stream>>>(
      h2, mlp_w1, mlp_b1, mlp_w2, mlp_b2, comb, mod, xmid, out);

  k_aux<<<1, 64, 0, stream>>>(prob, top1, out + BND);

  (void)in_sizes; (void)n_in; (void)out_size; (void)ws_size;
}